// CMoEGenerator_82248623718828
// MI455X (gfx1250) — compile-verified
//
#include <hip/hip_runtime.h>
#include <hip/hip_bf16.h>

#define B_  32
#define K_  8
#define S_  512
#define HS  256
#define ST  32          // rows of S per workgroup tile
#define PITCH 260       // LDS row pitch in floats (bank-conflict-free b64 A-frag reads)

typedef __attribute__((ext_vector_type(2))) float v2f;
typedef __attribute__((ext_vector_type(4))) float v4f;
typedef __attribute__((ext_vector_type(8))) float v8f;

// ---------------------------------------------------------------------------
// Kernel 1: streaming means over S.
//   blocks 0..B*(K+1)-1 ; last slot per b handles rem_fea, others ret_fea[k].
// ---------------------------------------------------------------------------
__global__ __launch_bounds__(256) void cmoe_means(const float* __restrict__ rem,
                                                  const float* __restrict__ ret,
                                                  float* __restrict__ mean_rem,
                                                  float* __restrict__ mean_ret) {
    const int blk = blockIdx.x;
    const int b   = blk / (K_ + 1);
    const int kk  = blk % (K_ + 1);
    const int d   = threadIdx.x;          // 0..255 == HS
    float s = 0.0f;
    if (kk == K_) {
        const float* p = rem + (size_t)b * S_ * HS + d;
        for (int ss = 0; ss < S_; ++ss) s += p[(size_t)ss * HS];
        mean_rem[b * HS + d] = s * (1.0f / (float)S_);
    } else {
        const float* p = ret + (((size_t)b * K_ + kk) * S_) * HS + d;
        for (int ss = 0; ss < S_; ++ss) s += p[(size_t)ss * HS];
        mean_ret[(b * K_ + kk) * HS + d] = s * (1.0f / (float)S_);
    }
}

// ---------------------------------------------------------------------------
// Kernel 2: avg_rem = mean_rem@Wr+br ; avg_ret = mean_ret@Wt+bt ;
//           scores -> softmax -> routing[B,K].   One block per b.
// ---------------------------------------------------------------------------
__global__ __launch_bounds__(256) void cmoe_router(const float* __restrict__ mean_rem,
                                                   const float* __restrict__ mean_ret,
                                                   const float* __restrict__ Wr,
                                                   const float* __restrict__ br,
                                                   const float* __restrict__ Wt,
                                                   const float* __restrict__ bt,
                                                   float* __restrict__ routing) {
    __shared__ float arem[HS];
    __shared__ float red[HS];
    __shared__ float sc[K_];
    const int b = blockIdx.x;
    const int e = threadIdx.x;            // 0..255 == HS

    // avg_rem[e]
    float acc = br[e];
    const float* mr = mean_rem + b * HS;
    for (int d = 0; d < HS; ++d) acc += mr[d] * Wr[d * HS + e];
    arem[e] = acc;
    __syncthreads();

    for (int k = 0; k < K_; ++k) {
        float a = bt[e];
        const float* mk = mean_ret + (b * K_ + k) * HS;
        for (int d = 0; d < HS; ++d) a += mk[d] * Wt[d * HS + e];
        red[e] = a * arem[e];
        __syncthreads();
        for (int st = 128; st > 0; st >>= 1) {
            if (e < st) red[e] += red[e + st];
            __syncthreads();
        }
        if (e == 0) sc[k] = red[0];
        __syncthreads();
    }
    if (e == 0) {
        float m = sc[0];
        for (int k = 1; k < K_; ++k) m = fmaxf(m, sc[k]);
        float ex[K_]; float sum = 0.0f;
        for (int k = 0; k < K_; ++k) { ex[k] = __expf(sc[k] - m); sum += ex[k]; }
        float inv = 1.0f / sum;
        for (int k = 0; k < K_; ++k) routing[b * K_ + k] = ex[k] * inv;
    }
}

// ---------------------------------------------------------------------------
// Kernel 3: experts via V_WMMA_F32_16X16X4_F32 with async LDS staging.
//   Workgroup = 8 waves, owns (b, 32-row S strip). Loops experts k,
//   accumulating out += relu(A@W1k+b1k) @ (r*W2k) + r*b2k in registers.
//   A tiles staged with GLOBAL_LOAD_ASYNC_TO_LDS_B128 (ASYNCcnt), next
//   expert's tile warmed into cache with GLOBAL_PREFETCH_B8.
// ---------------------------------------------------------------------------
__global__ __launch_bounds__(256) void cmoe_experts(const float* __restrict__ ret,
                                                    const float* __restrict__ W1,
                                                    const float* __restrict__ b1,
                                                    const float* __restrict__ W2,
                                                    const float* __restrict__ b2,
                                                    const float* __restrict__ routing,
                                                    float* __restrict__ out) {
    __shared__ float lds[ST * PITCH];     // 33280 B: A tile, then reused for h tile
    const int tid  = threadIdx.x;
    const int wave = tid >> 5;
    const int lane = tid & 31;
    const int half = lane >> 4;           // 0/1 : K-pair select per WMMA layout
    const int l16  = lane & 15;

    const int blk = blockIdx.x;           // 0 .. B*(S/ST)-1
    const int b   = blk / (S_ / ST);
    const int s0  = (blk % (S_ / ST)) * ST;

    const unsigned lds_base = (unsigned)(size_t)(&lds[0]);

    v8f out_acc[2][2] = {};               // [mtile][ntile] persistent over experts

    for (int k = 0; k < K_; ++k) {
        const float r = routing[b * K_ + k];
        const float* A_g = ret + (((size_t)b * K_ + k) * S_ + s0) * HS;  // 32x256 contiguous
        const float* W1k = W1 + (size_t)k * HS * HS;
        const float* W2k = W2 + (size_t)k * HS * HS;

        // ---- async-stage A tile into LDS (row-major, pitch 260) ----
        __syncthreads();                  // previous GEMM2 readers done
        {
            #pragma unroll
            for (int i = 0; i < 8; ++i) {
                int idx  = i * 256 + tid;          // b128 chunk index, 2048 total
                int row  = idx >> 6;               // 64 chunks per 256-float row
                int col4 = idx & 63;
                unsigned dst = lds_base + (unsigned)((row * PITCH + col4 * 4) * 4);
                unsigned long long ga = (unsigned long long)(size_t)(A_g + idx * 4);
                asm volatile("global_load_async_to_lds_b128 %0, %1, off"
                             :: "v"(dst), "v"(ga) : "memory");
            }
            // warm next expert's A tile into cache while this k computes
            if (k + 1 < K_) {
                const char* nxt =
                    (const char*)(ret + (((size_t)b * K_ + (k + 1)) * S_ + s0) * HS)
                    + (size_t)tid * 128;          // 256 threads x 128B = 32KB tile
                __builtin_prefetch(nxt, 0, 0);
            }
            asm volatile("s_wait_asynccnt 0x0" ::: "memory");
        }
        __syncthreads();

        // ---- GEMM1: h = relu(A @ W1k + b1k) ----
        v8f acc1[2][2] = {};
        for (int kk = 0; kk < HS; kk += 4) {
            v2f a0 = *(const v2f*)&lds[(0 * 16 + l16) * PITCH + kk + 2 * half];
            v2f a1 = *(const v2f*)&lds[(1 * 16 + l16) * PITCH + kk + 2 * half];
            #pragma unroll
            for (int nt = 0; nt < 2; ++nt) {
                const int n = wave * 32 + nt * 16 + l16;
                v2f bf;
                bf.x = W1k[(kk + 2 * half    ) * HS + n];
                bf.y = W1k[(kk + 2 * half + 1) * HS + n];
                acc1[0][nt] = __builtin_amdgcn_wmma_f32_16x16x4_f32(
                    false, a0, false, bf, (short)0, acc1[0][nt], false, false);
                acc1[1][nt] = __builtin_amdgcn_wmma_f32_16x16x4_f32(
                    false, a1, false, bf, (short)0, acc1[1][nt], false, false);
            }
        }
        __syncthreads();                  // all waves done reading A

        // bias + relu, write h into same LDS buffer
        #pragma unroll
        for (int nt = 0; nt < 2; ++nt) {
            const int n = wave * 32 + nt * 16 + l16;
            const float bias = b1[k * HS + n];
            #pragma unroll
            for (int mt = 0; mt < 2; ++mt) {
                #pragma unroll
                for (int rr = 0; rr < 8; ++rr) {
                    float h = fmaxf(acc1[mt][nt][rr] + bias, 0.0f);
                    const int row = mt * 16 + rr + 8 * half;    // C/D layout: M = r + 8*half
                    lds[row * PITCH + n] = h;
                }
            }
        }
        __syncthreads();

        // ---- GEMM2: out += h @ (r*W2k) ----
        for (int kk = 0; kk < HS; kk += 4) {
            v2f a0 = *(const v2f*)&lds[(0 * 16 + l16) * PITCH + kk + 2 * half];
            v2f a1 = *(const v2f*)&lds[(1 * 16 + l16) * PITCH + kk + 2 * half];
            #pragma unroll
            for (int nt = 0; nt < 2; ++nt) {
                const int n = wave * 32 + nt * 16 + l16;
                v2f bf;
                bf.x = r * W2k[(kk + 2 * half    ) * HS + n];
                bf.y = r * W2k[(kk + 2 * half + 1) * HS + n];
                out_acc[0][nt] = __builtin_amdgcn_wmma_f32_16x16x4_f32(
                    false, a0, false, bf, (short)0, out_acc[0][nt], false, false);
                out_acc[1][nt] = __builtin_amdgcn_wmma_f32_16x16x4_f32(
                    false, a1, false, bf, (short)0, out_acc[1][nt], false, false);
            }
        }
        // out += r * b2k
        #pragma unroll
        for (int nt = 0; nt < 2; ++nt) {
            const int n = wave * 32 + nt * 16 + l16;
            const float bias = r * b2[k * HS + n];
            #pragma unroll
            for (int mt = 0; mt < 2; ++mt)
                #pragma unroll
                for (int rr = 0; rr < 8; ++rr)
                    out_acc[mt][nt][rr] += bias;
        }
    }

    // ---- write gen_fea ----
    float* O = out + ((size_t)b * S_ + s0) * HS;
    #pragma unroll
    for (int mt = 0; mt < 2; ++mt) {
        #pragma unroll
        for (int nt = 0; nt < 2; ++nt) {
            const int n = wave * 32 + nt * 16 + l16;
            #pragma unroll
            for (int rr = 0; rr < 8; ++rr) {
                const int row = mt * 16 + rr + 8 * half;
                O[(size_t)row * HS + n] = out_acc[mt][nt][rr];
            }
        }
    }
}

// ---------------------------------------------------------------------------
extern "C" void kernel_launch(void* const* d_in, const int* in_sizes, int n_in,
                              void* d_out, int out_size, void* d_ws, size_t ws_size,
                              hipStream_t stream) {
    const float* rem = (const float*)d_in[0];
    const float* ret = (const float*)d_in[1];
    const float* Wr  = (const float*)d_in[2];
    const float* br  = (const float*)d_in[3];
    const float* Wt  = (const float*)d_in[4];
    const float* bt  = (const float*)d_in[5];
    const float* W1  = (const float*)d_in[6];
    const float* b1  = (const float*)d_in[7];
    const float* W2  = (const float*)d_in[8];
    const float* b2  = (const float*)d_in[9];
    float* out = (float*)d_out;

    float* ws       = (float*)d_ws;
    float* mean_rem = ws;                           // B*HS          = 8192 f
    float* mean_ret = ws + B_ * HS;                 // B*K*HS        = 65536 f
    float* routing  = ws + B_ * HS + B_ * K_ * HS;  // B*K           = 256 f
    (void)in_sizes; (void)n_in; (void)out_size; (void)ws_size;

    cmoe_means<<<B_ * (K_ + 1), 256, 0, stream>>>(rem, ret, mean_rem, mean_ret);
    cmoe_router<<<B_, 256, 0, stream>>>(mean_rem, mean_ret, Wr, br, Wt, bt, routing);
    cmoe_experts<<<B_ * (S_ / ST), 256, 0, stream>>>(ret, W1, b1, W2, b2, routing, out);
}